// DeltaMLP_2911987827332
// MI455X (gfx1250) — compile-verified
//
#include <hip/hip_runtime.h>
#include <hip/hip_bf16.h>

// Problem constants (from reference): B=4, N=512, D=128, H=128, C=3
constexpr int Bn = 4;
constexpr int Nn = 512;
constexpr int Dd = 128;
constexpr int Hh = 128;
constexpr int Cc = 3;
constexpr int Tt = Nn / 16;               // 32 j/i tiles per dim
constexpr int NPAIR = Tt * (Tt + 1) / 2;  // 528 tile pairs (ti <= tj)

typedef __attribute__((ext_vector_type(2))) float v2f;
typedef __attribute__((ext_vector_type(8))) float v8f;

// D(16x16,f32) = A(16x4,f32) x B(4x16,f32) + C  -- native f32 matrix path on CDNA5
__device__ __forceinline__ v8f wmma_f32_k4(v2f a, v2f b, v8f c) {
  // 8 args: (neg_a, A, neg_b, B, c_mod, C, reuse_a, reuse_b)
  return __builtin_amdgcn_wmma_f32_16x16x4_f32(false, a, false, b, (short)0, c,
                                               false, false);
}

__device__ __forceinline__ float silu_f(float x) {
  // x * sigmoid(x) = x / (1 + exp(-x)); fast rcp-based divide, v_exp_f32 path
  return __fdividef(x, 1.0f + __expf(-x));
}

// ---------------------------------------------------------------------------
// Kernel 1: projections  aP[m,h] = sum_d hd[m,d]*W1[h,d]      (+ b1 folded in)
//                        bP[m,h] = sum_d hd[m,d]*W1[h,D+d]
// m = b*N+n (2048 rows). grid = (128 m-tiles, 2 sel), 8 waves = 8 h-tiles.
// ---------------------------------------------------------------------------
__global__ void __launch_bounds__(256)
proj_kernel(const float* __restrict__ hd, const float* __restrict__ W1,
            const float* __restrict__ b1, float* __restrict__ aP,
            float* __restrict__ bP) {
  const int lane = threadIdx.x & 31;
  const int lh   = lane & 15;   // M (A) / N (B) index within tile
  const int hi   = lane >> 4;   // K half-select per f32 WMMA layout
  const int wave = threadIdx.x >> 5;

  const int m0  = blockIdx.x * 16;
  const int sel = blockIdx.y;       // 0 = src half of W1, 1 = dst half
  const int h0  = wave * 16;        // 8 waves cover H=128

  const float* arow = hd + (m0 + lh) * Dd;                 // A: row m, K=d
  const float* brow = W1 + (h0 + lh) * (2 * Dd) + sel * Dd; // B[k,n]=W1[h0+n, d=k]

  v8f acc = {};
  #pragma unroll
  for (int kk = 0; kk < Dd / 4; ++kk) {
    const int d = 4 * kk + 2 * hi;               // K pair {d, d+1}
    v2f afrag = *(const v2f*)(arow + d);
    v2f bfrag = *(const v2f*)(brow + d);
    acc = wmma_f32_k4(afrag, bfrag, acc);
  }

  float* P = sel ? bP : aP;
  const int h = h0 + lh;
  const float bias = (sel == 0) ? b1[h] : 0.0f;  // fold b1 into src projection
  #pragma unroll
  for (int r = 0; r < 8; ++r) {
    const int m = m0 + r + 8 * hi;               // C/D layout: M = r (+8 hi-half)
    P[m * Hh + h] = acc[r] + bias;
  }
}

// ---------------------------------------------------------------------------
// Kernel 2: fused SiLU + (1Mx128)@(128x3) contraction + symmetrization.
// One block per (batch, tile-pair ti<=tj). LDS holds aProj/bProj rows for both
// tiles (stride 132 floats -> conflict-free), padded W2^T, and both 16x16x3
// logits tiles so the symmetrized output is written directly (no HBM staging).
// ---------------------------------------------------------------------------
__global__ void __launch_bounds__(256)
pair_kernel(const float* __restrict__ aP, const float* __restrict__ bP,
            const float* __restrict__ W2, const float* __restrict__ b2,
            float* __restrict__ out) {
  __shared__ float sAi[16][132];   // aProj rows of i-tile (b1 folded)
  __shared__ float sBi[16][132];   // bProj rows of i-tile
  __shared__ float sAj[16][132];   // aProj rows of j-tile
  __shared__ float sBj[16][132];   // bProj rows of j-tile
  __shared__ float sW2[16][132];   // W2 padded 3->16 "channel" rows
  __shared__ float L1[16][16][4];  // logits[i-rows, j-cols, c]
  __shared__ float L2[16][16][4];  // logits[j-rows, i-cols, c]

  const int tid  = threadIdx.x;
  const int lane = tid & 31;
  const int lh   = lane & 15;
  const int hi   = lane >> 4;
  const int wave = tid >> 5;

  // decode linear tile-pair -> (ti, tj) with ti <= tj (uniform SALU loop)
  int tp = blockIdx.x, ti = 0, row = Tt;
  while (tp >= row) { tp -= row; --row; ++ti; }
  const int tj = ti + tp;
  const int b  = blockIdx.y;
  const int i0 = ti * 16, j0 = tj * 16;

  // ---- stage LDS ----
  for (int idx = tid; idx < 16 * Hh; idx += 256) {
    const int r = idx >> 7, h = idx & (Hh - 1);
    sAi[r][h] = aP[(b * Nn + i0 + r) * Hh + h];
    sBi[r][h] = bP[(b * Nn + i0 + r) * Hh + h];
    sAj[r][h] = aP[(b * Nn + j0 + r) * Hh + h];
    sBj[r][h] = bP[(b * Nn + j0 + r) * Hh + h];
    sW2[r][h] = (r < Cc) ? W2[r * Hh + h] : 0.0f;
  }
  __syncthreads();

  // ---- 32 row-tasks: 16 rows of orientation 1, 16 of orientation 2 ----
  // task t: orient = t>>4, local row = t&15. Each wave runs 4 tasks.
  for (int s = 0; s < 4; ++s) {
    const int t      = wave * 4 + s;
    const int orient = t >> 4;
    const int rr     = t & 15;
    const float(*rowA)[132] = orient ? sAj : sAi;  // fixed row source
    const float(*colB)[132] = orient ? sBi : sBj;  // 16 partner rows (A's M dim)

    v8f acc = {};
    #pragma unroll 4
    for (int kk = 0; kk < Hh / 4; ++kk) {
      const int h = 4 * kk + 2 * hi;
      // A[m,h] = silu(aProj[row] + bProj[partner m] (+b1 already folded))
      const float x0 = rowA[rr][h]     + colB[lh][h];
      const float x1 = rowA[rr][h + 1] + colB[lh][h + 1];
      v2f afrag;
      afrag[0] = silu_f(x0);
      afrag[1] = silu_f(x1);
      v2f bfrag = *(const v2f*)&sW2[lh][h];        // B[k,n] = W2pad[n,h=k]
      acc = wmma_f32_k4(afrag, bfrag, acc);
    }

    float(*Lb)[16][4] = orient ? L2 : L1;
    #pragma unroll
    for (int r = 0; r < 8; ++r) {
      if (lh < Cc) Lb[rr][r + 8 * hi][lh] = acc[r];  // (M=partner, N=c)
    }
  }
  __syncthreads();

  // ---- symmetrize + write both mirrored tiles ----
  const float b2v[3] = {b2[0], b2[1], b2[2]};
  for (int idx = tid; idx < 16 * 16 * Cc; idx += 256) {
    const int x = idx / (16 * Cc);
    const int r = idx % (16 * Cc);
    const int y = r / Cc;
    const int c = r % Cc;
    const float v = 0.5f * (L1[x][y][c] + L2[y][x][c]) + b2v[c];
    out[((b * Nn + (i0 + x)) * Nn + (j0 + y)) * Cc + c] = v;
    out[((b * Nn + (j0 + y)) * Nn + (i0 + x)) * Cc + c] = v;  // diag: same val
  }
}

extern "C" void kernel_launch(void* const* d_in, const int* in_sizes, int n_in,
                              void* d_out, int out_size, void* d_ws,
                              size_t ws_size, hipStream_t stream) {
  (void)in_sizes; (void)n_in; (void)out_size; (void)ws_size;
  const float* hd = (const float*)d_in[0];  // (B,N,D)
  const float* W1 = (const float*)d_in[1];  // (H,2D)
  const float* b1 = (const float*)d_in[2];  // (H)
  const float* W2 = (const float*)d_in[3];  // (C,H)
  const float* b2 = (const float*)d_in[4];  // (C)
  float* out = (float*)d_out;               // (B,N,N,C)

  float* aP = (float*)d_ws;                 // (B*N, H) src proj (+b1)
  float* bP = aP + Bn * Nn * Hh;            // (B*N, H) dst proj   (2 MB total)

  proj_kernel<<<dim3((Bn * Nn) / 16, 2), 256, 0, stream>>>(hd, W1, b1, aP, bP);
  pair_kernel<<<dim3(NPAIR, Bn), 256, 0, stream>>>(aP, bP, W2, b2, out);
}